// MemoEnhancedPredictor_54571854463676
// MI455X (gfx1250) — compile-verified
//
#include <hip/hip_runtime.h>
#include <cstddef>
#include <cstdint>

#define THREADS 256
#define NBATCH_BLOCKS 64
#define NPART (4 + NBATCH_BLOCKS)

typedef __attribute__((ext_vector_type(2))) float v2f;
typedef __attribute__((ext_vector_type(4))) float v4f;
typedef __attribute__((ext_vector_type(8))) float v8f;

// ---------------------------------------------------------------------------
// Kernel 1: row softmax (C=4), entropy, argmax (first-max like jnp.argmax)
// ---------------------------------------------------------------------------
__global__ void mep_k1_softmax(const float* __restrict__ logits,
                               float* __restrict__ pred,
                               float* __restrict__ ent,
                               int* __restrict__ pseudo,
                               int B) {
  int r = blockIdx.x * blockDim.x + threadIdx.x;
  if (r >= B) return;
  float4 l = ((const float4*)logits)[r];
  float mx = fmaxf(fmaxf(l.x, l.y), fmaxf(l.z, l.w));
  float d0 = l.x - mx, d1 = l.y - mx, d2 = l.z - mx, d3 = l.w - mx;
  float e0 = expf(d0), e1 = expf(d1), e2 = expf(d2), e3 = expf(d3);
  float s = e0 + e1 + e2 + e3;
  float inv = 1.0f / s;
  ((float4*)pred)[r] = make_float4(e0 * inv, e1 * inv, e2 * inv, e3 * inv);
  float dot = e0 * d0 + e1 * d1 + e2 * d2 + e3 * d3;
  ent[r] = logf(s) - dot * inv;
  int am = 0; float bm = l.x;
  if (l.y > bm) { bm = l.y; am = 1; }
  if (l.z > bm) { bm = l.z; am = 2; }
  if (l.w > bm) { bm = l.w; am = 3; }
  pseudo[r] = am;
}

// entropies are >= 0 (clamped), so float bits compare monotonically as uint
__device__ __forceinline__ unsigned mep_key(float e) {
  return __float_as_uint(fmaxf(e, 0.0f));
}

// ---------------------------------------------------------------------------
// Kernel 2: per-label exact top-512-smallest selection via MSB radix select.
// One block per label. Ties at the threshold resolved in stable-argsort order
// (memo indices first, then batch index order).
// ---------------------------------------------------------------------------
__global__ void mep_k2_select(const float* __restrict__ ent_memo, // [4*M]
                              const float* __restrict__ ent,      // [B]
                              const int* __restrict__ pseudo,     // [B]
                              int* __restrict__ sel_memo,         // [4*M]
                              int* __restrict__ sel_batch,        // [B]
                              int B, int M) {
  const int c = blockIdx.x;
  const int t = threadIdx.x;
  __shared__ unsigned red[THREADS];
  __shared__ unsigned s_prefix;
  __shared__ int s_want;
  const unsigned INFK = 0x7f800000u; // +inf
  if (t == 0) { s_prefix = 0u; s_want = M; }
  __syncthreads();

  for (int bit = 31; bit >= 0; --bit) {
    unsigned pref0 = s_prefix << 1; // bits [31:bit] with current bit = 0
    unsigned cnt = 0;
    for (int m = t; m < M; m += THREADS) {
      unsigned k = mep_key(ent_memo[c * M + m]);
      cnt += ((k >> bit) == pref0);
    }
    for (int b = t; b < B; b += THREADS) {
      unsigned k = (pseudo[b] == c) ? mep_key(ent[b]) : INFK;
      cnt += ((k >> bit) == pref0);
    }
    red[t] = cnt;
    __syncthreads();
    for (int off = THREADS / 2; off > 0; off >>= 1) {
      if (t < off) red[t] += red[t + off];
      __syncthreads();
    }
    if (t == 0) {
      unsigned c0 = red[0];
      if ((unsigned)s_want <= c0) s_prefix = pref0;
      else { s_want -= (int)c0; s_prefix = pref0 | 1u; }
    }
    __syncthreads();
  }
  const unsigned T = s_prefix; // key of the M-th smallest value

  // count strictly-less, then flag
  unsigned cnt = 0;
  for (int m = t; m < M; m += THREADS) cnt += (mep_key(ent_memo[c * M + m]) < T);
  for (int b = t; b < B; b += THREADS)
    if (pseudo[b] == c) cnt += (mep_key(ent[b]) < T);
  red[t] = cnt;
  __syncthreads();
  for (int off = THREADS / 2; off > 0; off >>= 1) {
    if (t < off) red[t] += red[t + off];
    __syncthreads();
  }

  for (int m = t; m < M; m += THREADS)
    sel_memo[c * M + m] = (mep_key(ent_memo[c * M + m]) < T) ? 1 : 0;
  for (int b = t; b < B; b += THREADS)
    if (pseudo[b] == c) sel_batch[b] = (mep_key(ent[b]) < T) ? 1 : 0;
  __syncthreads();

  if (t == 0) {
    int q = M - (int)red[0]; // how many threshold-equal items to include
    int got = 0;
    for (int m = 0; m < M && got < q; ++m)
      if (mep_key(ent_memo[c * M + m]) == T) { sel_memo[c * M + m] = 1; ++got; }
    for (int b = 0; b < B && got < q; ++b)
      if (pseudo[b] == c && mep_key(ent[b]) == T) { sel_batch[b] = 1; ++got; }
  }
}

// ---------------------------------------------------------------------------
// Kernel 3: deterministic partial sums of selected text embeddings.
// Blocks 0..3: memo rows of label c. Blocks 4..: batch chunks.
// D must equal 2*THREADS (=512). Partials fully written each launch.
// ---------------------------------------------------------------------------
__global__ void mep_k3_partials(const float* __restrict__ text_memo, // [4*M*D]
                                const float* __restrict__ text_emb,  // [B*D]
                                const int* __restrict__ sel_memo,
                                const int* __restrict__ sel_batch,
                                const int* __restrict__ pseudo,
                                float* __restrict__ partial,         // [NPART*4*D]
                                int B, int M, int D) {
  const int p = blockIdx.x;
  const int t = threadIdx.x;
  float a00 = 0, a01 = 0, a10 = 0, a11 = 0, a20 = 0, a21 = 0, a30 = 0, a31 = 0;
  if (p < 4) {
    const int c = p;
    const float* base = text_memo + (size_t)c * M * D;
    for (int m = 0; m < M; ++m) {
      if (sel_memo[c * M + m]) { // uniform branch per row
        float v0 = base[(size_t)m * D + t];
        float v1 = base[(size_t)m * D + t + THREADS];
        if      (c == 0) { a00 += v0; a01 += v1; }
        else if (c == 1) { a10 += v0; a11 += v1; }
        else if (c == 2) { a20 += v0; a21 += v1; }
        else             { a30 += v0; a31 += v1; }
      }
    }
  } else {
    const int per = B / NBATCH_BLOCKS;
    const int b0 = (p - 4) * per, b1 = b0 + per;
    for (int b = b0; b < b1; ++b) {
      if (sel_batch[b]) { // uniform per item: unselected items skip their loads
        int lbl = pseudo[b];
        float v0 = text_emb[(size_t)b * D + t];
        float v1 = text_emb[(size_t)b * D + t + THREADS];
        if      (lbl == 0) { a00 += v0; a01 += v1; }
        else if (lbl == 1) { a10 += v0; a11 += v1; }
        else if (lbl == 2) { a20 += v0; a21 += v1; }
        else               { a30 += v0; a31 += v1; }
      }
    }
  }
  float* out = partial + (size_t)p * 4 * D;
  out[0 * D + t] = a00; out[0 * D + t + THREADS] = a01;
  out[1 * D + t] = a10; out[1 * D + t + THREADS] = a11;
  out[2 * D + t] = a20; out[2 * D + t + THREADS] = a21;
  out[3 * D + t] = a30; out[3 * D + t + THREADS] = a31;
}

// ---------------------------------------------------------------------------
// Kernel 4: reduce partials -> Spad[16][D]; rows 4..15 zeroed (N padding).
// ---------------------------------------------------------------------------
__global__ void mep_k4_reduceS(const float* __restrict__ partial,
                               float* __restrict__ Spad, // [16*D]
                               int D) {
  int g = blockIdx.x * blockDim.x + threadIdx.x; // 0 .. 16*D-1
  if (g < 4 * D) {
    float s = 0.f;
    for (int p = 0; p < NPART; ++p) s += partial[(size_t)p * 4 * D + g];
    Spad[g] = s;
  } else {
    Spad[g] = 0.f;
  }
}

// ---------------------------------------------------------------------------
// Kernel 5: skinny GEMM cosin[b,n] = sum_k text_emb[b,k]*Spad[n,k] with
// v_wmma_f32_16x16x4_f32, fused 2-way-softmax epilogue -> memo_pred.
// One wave per 16 batch rows; Spad staged in LDS (32 KB).
// ---------------------------------------------------------------------------
__global__ void __launch_bounds__(32)
mep_k5_gemm(const float* __restrict__ text_emb,
            const float* __restrict__ Spad,   // [16*512]
            float* __restrict__ memo_pred,    // [B*4]
            int D) {
  __shared__ float sB[16 * 512];
  __shared__ float sC[16 * 4];
  const int lane = threadIdx.x;

  // stage B matrix (S padded to 16 rows) into LDS
  {
    const v4f* src = (const v4f*)Spad;
    v4f* dst = (v4f*)sB;
#pragma unroll
    for (int i = 0; i < 64; ++i) dst[i * 32 + lane] = src[i * 32 + lane];
  }
  __syncthreads();

  const int rowbase = blockIdx.x * 16;
  const int idx16 = lane & 15;   // A: M index / B: N index / C: N index
  const int khalf = lane >> 4;   // selects K pair {0,1} vs {2,3} of each step
  const float* arow = text_emb + (size_t)(rowbase + idx16) * D;

  v8f acc = {};
  for (int k = 0; k < D; k += 4) {
    const int kk = k + khalf * 2;
    v2f a = *(const v2f*)(arow + kk);                // A[m][kk], A[m][kk+1]
    v2f b = *(const v2f*)(sB + idx16 * 512 + kk);    // B[kk][n], B[kk+1][n] = S[n][kk..]
    acc = __builtin_amdgcn_wmma_f32_16x16x4_f32(
        /*neg_a=*/false, a, /*neg_b=*/false, b,
        /*c_mod=*/(short)0, acc, /*reuse_a=*/false, /*reuse_b=*/false);
  }

  // C/D layout: lane holds (m = r + 8*khalf, n = idx16), value acc[r]
  if (idx16 < 4) {
#pragma unroll
    for (int r = 0; r < 8; ++r) sC[(r + 8 * khalf) * 4 + idx16] = acc[r];
  }
  __syncthreads();

  if (lane < 16) {
    float ct0 = sC[lane * 4 + 0], ct1 = sC[lane * 4 + 1];
    float ct2 = sC[lane * 4 + 2], ct3 = sC[lane * 4 + 3];
    float tx0 = ct0 + ct2, tx1 = ct1 + ct3;       // text_combine
    float vx0 = ct0 + ct1, vx1 = ct2 + ct3;       // vision_combine (uses ct, per ref)
    float tm = fmaxf(tx0, tx1);
    float te0 = expf(tx0 - tm), te1 = expf(tx1 - tm);
    float tis = 1.0f / (te0 + te1);
    float t0 = te0 * tis, t1 = te1 * tis;
    float vm = fmaxf(vx0, vx1);
    float ve0 = expf(vx0 - vm), ve1 = expf(vx1 - vm);
    float vis = 1.0f / (ve0 + ve1);
    float v0 = ve0 * vis, v1 = ve1 * vis;
    ((float4*)memo_pred)[rowbase + lane] =
        make_float4(t0 * v0, t1 * v0, t0 * v1, t1 * v1);
  }
}

// ---------------------------------------------------------------------------
extern "C" void kernel_launch(void* const* d_in, const int* in_sizes, int n_in,
                              void* d_out, int out_size, void* d_ws, size_t ws_size,
                              hipStream_t stream) {
  const float* logits    = (const float*)d_in[0];
  const float* text_emb  = (const float*)d_in[1];
  // d_in[2] vision_embeds: dead code w.r.t. the returned outputs
  const float* ent_memo  = (const float*)d_in[3];
  const float* text_memo = (const float*)d_in[4];
  // d_in[5] vision_memo: dead code w.r.t. the returned outputs

  const int C = 4;
  const int B = in_sizes[0] / C;   // 16384
  const int D = in_sizes[1] / B;   // 512
  const int M = in_sizes[3] / C;   // 512

  float* out = (float*)d_out;
  float* memo_pred = out;                    // [B,4]
  float* pred      = out + (size_t)B * C;    // [B,4]
  float* ent       = out + (size_t)2 * B * C;// [B]

  char* ws = (char*)d_ws;
  int* pseudo    = (int*)ws;  ws += (size_t)B * sizeof(int);
  int* sel_batch = (int*)ws;  ws += (size_t)B * sizeof(int);
  int* sel_memo  = (int*)ws;  ws += (size_t)C * M * sizeof(int);
  float* partial = (float*)ws; ws += (size_t)NPART * 4 * D * sizeof(float);
  float* Spad    = (float*)ws; // [16*D]

  mep_k1_softmax<<<(B + THREADS - 1) / THREADS, THREADS, 0, stream>>>(
      logits, pred, ent, pseudo, B);
  mep_k2_select<<<4, THREADS, 0, stream>>>(
      ent_memo, ent, pseudo, sel_memo, sel_batch, B, M);
  mep_k3_partials<<<NPART, THREADS, 0, stream>>>(
      text_memo, text_emb, sel_memo, sel_batch, pseudo, partial, B, M, D);
  mep_k4_reduceS<<<(16 * D) / THREADS, THREADS, 0, stream>>>(partial, Spad, D);
  mep_k5_gemm<<<B / 16, 32, 0, stream>>>(text_emb, Spad, memo_pred, D);
}